// GATClassifier_26302379720854
// MI455X (gfx1250) — compile-verified
//
#include <hip/hip_runtime.h>
#include <math.h>

// ---------------- problem constants (match reference) ----------------
#define N_NODES   30000
#define N_EDGES   480000
#define N_GRAPHS  256
#define F_INP     64
#define HIDC      32            // channels per head
#define HEADS4    4
#define HC128     (HEADS4*HIDC) // 128
#define EPRIME    (N_EDGES + N_NODES)   // edges + self loops = 510000
#define SLOPE_ATT 0.2f
#define SLOPE_ACT 0.01f

typedef __attribute__((ext_vector_type(2))) float v2f;
typedef __attribute__((ext_vector_type(8))) float v8f;

__device__ __forceinline__ float lrelu(float v, float s) { return v > 0.f ? v : v * s; }

// =====================================================================
// fp32 WMMA GEMM: C[M,Nout] = A[M,K] x B[K,Nout]
// One wave32 per 16x16 output tile; K stepped by 4 via V_WMMA_F32_16X16X4_F32.
// M=30000 (=1875*16), K in {64,128}, Nout in {128,32}: all exact multiples,
// so no bounds checks -> EXEC stays all-ones at every wmma (ISA requirement).
// A frag (16x4): lane&15 = row M, lane>>4 selects K pair {0,1}/{2,3}.
// B frag (4x16): lane&15 = col N, lane>>4 selects K pair (symmetric layout).
// C/D (16x16 f32): vgpr v -> M = v + 8*(lane>>4), N = lane&15.
// =====================================================================
__global__ void gemm_wmma_f32(const float* __restrict__ A,
                              const float* __restrict__ B,
                              float* __restrict__ C,
                              int K, int Nout) {
  const int mBase = blockIdx.x * 16;
  const int nBase = blockIdx.y * 16;
  const int lane  = threadIdx.x;      // 0..31, wave32
  const int half  = lane >> 4;        // 0 or 1
  const int l15   = lane & 15;

  v8f acc = {};
  const float* Arow = A + (size_t)(mBase + l15) * K;
  const float* Bcol = B + nBase + l15;

  for (int k = 0; k < K; k += 4) {
    const int ka = k + half * 2;
    v2f a, b;
    a.x = Arow[ka];
    a.y = Arow[ka + 1];
    b.x = Bcol[(size_t)(ka)     * Nout];
    b.y = Bcol[(size_t)(ka + 1) * Nout];
    acc = __builtin_amdgcn_wmma_f32_16x16x4_f32(
        /*neg_a=*/false, a, /*neg_b=*/false, b,
        /*c_mod=*/(short)0, acc, /*reuse_a=*/false, /*reuse_b=*/false);
  }

#pragma unroll
  for (int v = 0; v < 8; ++v) {
    C[(size_t)(mBase + v + half * 8) * Nout + nBase + l15] = acc[v];
  }
}

// ---------------- per-node attention logits: al = <h[n,h,:], a[h,:]> -----
__global__ void attn_logits(const float* __restrict__ h,
                            const float* __restrict__ a_src,
                            const float* __restrict__ a_dst,
                            float* __restrict__ alS,
                            float* __restrict__ alD,
                            int Hn) {
  int t = blockIdx.x * blockDim.x + threadIdx.x;
  if (t >= N_NODES * Hn) return;
  const int n  = t / Hn;
  const int hh = t - n * Hn;
  const float* hp = h + ((size_t)n * Hn + hh) * HIDC;
  const float* as = a_src + hh * HIDC;
  const float* ad = a_dst + hh * HIDC;
  float s = 0.f, d = 0.f;
#pragma unroll
  for (int c = 0; c < HIDC; ++c) { float v = hp[c]; s += v * as[c]; d += v * ad[c]; }
  alS[t] = s;
  alD[t] = d;
}

// ---------------- workspace fill ----------------
__global__ void fill_f32(float* __restrict__ p, float v, int n) {
  int t = blockIdx.x * blockDim.x + threadIdx.x;
  if (t < n) p[t] = v;
}

__device__ __forceinline__ void edge_sd(const int* __restrict__ ei, int e,
                                        int& s, int& d) {
  if (e < N_EDGES) { s = ei[e]; d = ei[N_EDGES + e]; }
  else             { s = d = e - N_EDGES; }       // implicit self loop
}

// ---------------- edge pass 1: segment max of leaky attention ----------
__global__ void edge_attn_max(const int* __restrict__ ei,
                              const float* __restrict__ alS,
                              const float* __restrict__ alD,
                              float* __restrict__ m, int Hn) {
  int e = blockIdx.x * blockDim.x + threadIdx.x;
  if (e >= EPRIME) return;
  int s, d; edge_sd(ei, e, s, d);
  for (int hh = 0; hh < Hn; ++hh) {
    float v = lrelu(alS[s * Hn + hh] + alD[d * Hn + hh], SLOPE_ATT);
    atomicMax(&m[d * Hn + hh], v);
  }
}

// ---------------- edge pass 2: exp numerators + segment denominator ----
__global__ void edge_attn_exp(const int* __restrict__ ei,
                              const float* __restrict__ alS,
                              const float* __restrict__ alD,
                              const float* __restrict__ m,
                              float* __restrict__ denom,
                              float* __restrict__ ee, int Hn) {
  int e = blockIdx.x * blockDim.x + threadIdx.x;
  if (e >= EPRIME) return;
  int s, d; edge_sd(ei, e, s, d);
  for (int hh = 0; hh < Hn; ++hh) {
    float v = lrelu(alS[s * Hn + hh] + alD[d * Hn + hh], SLOPE_ATT);
    float w = __expf(v - m[d * Hn + hh]);
    ee[(size_t)e * Hn + hh] = w;
    atomicAdd(&denom[d * Hn + hh], w);
  }
}

// ---------------- edge pass 3: alpha-weighted message scatter-add ------
__global__ void edge_message(const int* __restrict__ ei,
                             const float* __restrict__ ee,
                             const float* __restrict__ denom,
                             const float* __restrict__ h,
                             float* __restrict__ accum, int Hn) {
  int t = blockIdx.x * blockDim.x + threadIdx.x;
  if (t >= EPRIME * Hn) return;
  const int e  = t / Hn;
  const int hh = t - e * Hn;
  int s, d; edge_sd(ei, e, s, d);
  float alpha = ee[t] / (denom[d * Hn + hh] + 1e-16f);
  const float* hs = h + ((size_t)s * Hn + hh) * HIDC;
  float* ap = accum + ((size_t)d * Hn + hh) * HIDC;
#pragma unroll
  for (int c = 0; c < HIDC; ++c) atomicAdd(&ap[c], alpha * hs[c]);
}

// ---------------- bias + (leaky) activation, in place ------------------
__global__ void bias_act(float* __restrict__ x, const float* __restrict__ b,
                         int HCn, float slope, int total) {
  int t = blockIdx.x * blockDim.x + threadIdx.x;
  if (t >= total) return;
  x[t] = lrelu(x[t] + b[t % HCn], slope);   // slope==1 -> identity
}

// ---------------- global mean pool (accumulate) ------------------------
__global__ void pool_accum(const float* __restrict__ h3,
                           const int* __restrict__ batch,
                           float* __restrict__ sums,
                           float* __restrict__ cnt) {
  int n = blockIdx.x * blockDim.x + threadIdx.x;
  if (n >= N_NODES) return;
  int g = batch[n];
  const float* hp = h3 + (size_t)n * HIDC;
#pragma unroll
  for (int c = 0; c < HIDC; ++c) atomicAdd(&sums[g * HIDC + c], hp[c]);
  atomicAdd(&cnt[g], 1.0f);
}

// ---------------- final linear + 2-way softmax -------------------------
__global__ void head_kernel(const float* __restrict__ sums,
                            const float* __restrict__ cnt,
                            const float* __restrict__ Wl,
                            const float* __restrict__ bl,
                            float* __restrict__ out) {
  int g = blockIdx.x * blockDim.x + threadIdx.x;
  if (g >= N_GRAPHS) return;
  float inv = 1.0f / fmaxf(cnt[g], 1.0f);
  float l0 = bl[0], l1 = bl[1];
#pragma unroll
  for (int c = 0; c < HIDC; ++c) {
    float p = sums[g * HIDC + c] * inv;
    l0 += p * Wl[c * 2 + 0];
    l1 += p * Wl[c * 2 + 1];
  }
  float mx = fmaxf(l0, l1);
  float e0 = __expf(l0 - mx), e1 = __expf(l1 - mx);
  float s = e0 + e1;
  out[g * 2 + 0] = e0 / s;
  out[g * 2 + 1] = e1 / s;
}

// =====================================================================
// host side
// =====================================================================
static void run_gat_layer(const float* X, int K,
                          const float* W, const float* as_, const float* ad_,
                          const float* b_, int Hn, float act_slope,
                          const int* ei,
                          float* hbuf, float* accum,
                          float* alS, float* alD, float* m, float* den,
                          float* ee, hipStream_t stream) {
  const int HCn = Hn * HIDC;
  dim3 ggrid(N_NODES / 16, HCn / 16);
  gemm_wmma_f32<<<ggrid, 32, 0, stream>>>(X, W, hbuf, K, HCn);

  const int nh = N_NODES * Hn;
  attn_logits<<<(nh + 255) / 256, 256, 0, stream>>>(hbuf, as_, ad_, alS, alD, Hn);

  fill_f32<<<(nh + 255) / 256, 256, 0, stream>>>(m, -1e30f, nh);
  fill_f32<<<(nh + 255) / 256, 256, 0, stream>>>(den, 0.f, nh);
  const int tot = N_NODES * HCn;
  fill_f32<<<(tot + 255) / 256, 256, 0, stream>>>(accum, 0.f, tot);

  edge_attn_max<<<(EPRIME + 255) / 256, 256, 0, stream>>>(ei, alS, alD, m, Hn);
  edge_attn_exp<<<(EPRIME + 255) / 256, 256, 0, stream>>>(ei, alS, alD, m, den, ee, Hn);
  const int eh = EPRIME * Hn;
  edge_message<<<(eh + 255) / 256, 256, 0, stream>>>(ei, ee, den, hbuf, accum, Hn);

  bias_act<<<(tot + 255) / 256, 256, 0, stream>>>(accum, b_, HCn, act_slope, tot);
}

extern "C" void kernel_launch(void* const* d_in, const int* in_sizes, int n_in,
                              void* d_out, int out_size, void* d_ws, size_t ws_size,
                              hipStream_t stream) {
  (void)in_sizes; (void)n_in; (void)out_size; (void)ws_size;
  const float* x     = (const float*)d_in[0];
  const int*   ei    = (const int*)  d_in[1];
  const int*   batch = (const int*)  d_in[2];
  const float* W1  = (const float*)d_in[3];
  const float* as1 = (const float*)d_in[4];
  const float* ad1 = (const float*)d_in[5];
  const float* b1  = (const float*)d_in[6];
  const float* W2  = (const float*)d_in[7];
  const float* as2 = (const float*)d_in[8];
  const float* ad2 = (const float*)d_in[9];
  const float* b2  = (const float*)d_in[10];
  const float* W3  = (const float*)d_in[11];
  const float* as3 = (const float*)d_in[12];
  const float* ad3 = (const float*)d_in[13];
  const float* b3  = (const float*)d_in[14];
  const float* Wl  = (const float*)d_in[15];
  const float* bl  = (const float*)d_in[16];
  float* out = (float*)d_out;

  // workspace carve-up (floats); total ~10.2M floats ~= 41 MB
  float* ws  = (float*)d_ws;
  float* hA  = ws;                                    // N * 128
  float* hB  = hA  + (size_t)N_NODES * HC128;         // N * 128
  float* alS = hB  + (size_t)N_NODES * HC128;         // N * HEADS
  float* alD = alS + (size_t)N_NODES * HEADS4;
  float* m   = alD + (size_t)N_NODES * HEADS4;
  float* den = m   + (size_t)N_NODES * HEADS4;
  float* ee  = den + (size_t)N_NODES * HEADS4;        // EPRIME * HEADS
  float* sums= ee  + (size_t)EPRIME  * HEADS4;        // G * HID
  float* cnt = sums + (size_t)N_GRAPHS * HIDC;        // G

  // Layer 1: x[N,64] -> hA[N,128]; aggregate into hB; leaky 0.01
  run_gat_layer(x, F_INP, W1, as1, ad1, b1, HEADS4, SLOPE_ACT,
                ei, hA, hB, alS, alD, m, den, ee, stream);
  // Layer 2: hB[N,128] -> hA[N,128]; aggregate back into hB; leaky 0.01
  run_gat_layer(hB, HC128, W2, as2, ad2, b2, HEADS4, SLOPE_ACT,
                ei, hA, hB, alS, alD, m, den, ee, stream);
  // Layer 3: hB[N,128] -> hA[N,32] (1 head); aggregate into hB[N,32]; no act
  run_gat_layer(hB, HC128, W3, as3, ad3, b3, 1, 1.0f,
                ei, hA, hB, alS, alD, m, den, ee, stream);

  // global mean pool + linear head + softmax
  const int gs = N_GRAPHS * HIDC;
  fill_f32<<<(gs + 255) / 256, 256, 0, stream>>>(sums, 0.f, gs);
  fill_f32<<<(N_GRAPHS + 255) / 256, 256, 0, stream>>>(cnt, 0.f, N_GRAPHS);
  pool_accum<<<(N_NODES + 255) / 256, 256, 0, stream>>>(hB, batch, sums, cnt);
  head_kernel<<<1, 256, 0, stream>>>(sums, cnt, Wl, bl, out);
}